// MultiInjection_75196287418420
// MI455X (gfx1250) — compile-verified
//
#include <hip/hip_runtime.h>
#include <hip/hip_bf16.h>
#include <math.h>

typedef float v2f __attribute__((ext_vector_type(2)));
typedef float v4f __attribute__((ext_vector_type(4)));
typedef float v8f __attribute__((ext_vector_type(8)));

#if defined(__HIP_DEVICE_COMPILE__) && !__has_builtin(__builtin_amdgcn_wmma_f32_16x16x4_f32)
#error "gfx1250 device pass: __builtin_amdgcn_wmma_f32_16x16x4_f32 not available"
#endif

__device__ __forceinline__ v8f wmma4(v2f a, v2f b, v8f c) {
  // 8-arg form: (neg_a, A, neg_b, B, c_mod, C, reuse_a, reuse_b)
  return __builtin_amdgcn_wmma_f32_16x16x4_f32(false, a, false, b, (short)0, c, false, false);
}

#define ACT_NONE 0
#define ACT_TANH 1
#define ACT_RELU 2
#define ACT_SIGM 3

// ---------------------------------------------------------------------------
// Generic tiled WMMA f32 GEMM: C = act((A [+A2]) @ B + bias) [* Mul]
// Row-major A[M,K], B[K,N], C[M,N]. Batched over blockIdx.z with strides.
// Block tile 64x64, 4 waves (128 thr), wave tile 32x32 (2x2 WMMA frags).
// ---------------------------------------------------------------------------
#define BM 64
#define BN 64
#define BK 16
#define LSTR 20   // padded LDS row stride (floats), keeps 16B alignment + bank spread

__global__ __launch_bounds__(128)
void gemm_wmma_f32(const float* A, const float* A2, const float* B,
                   const float* bias, const float* Mul, float* C,
                   int M, int N, int K,
                   long strideA, long strideB, long strideC, int act)
{
  __shared__ float Al[BM * LSTR];
  __shared__ float Bl[BN * LSTR];   // stored transposed: Bl[n*LSTR + k]

  const long z = blockIdx.z;
  A += z * strideA;
  if (A2) A2 += z * strideA;
  B += z * strideB;
  C += z * strideC;
  const float* MulP = Mul ? (Mul + z * strideC) : (const float*)0;

  const int tid  = threadIdx.x;
  const int lane = tid & 31;
  const int wv   = tid >> 5;          // wave 0..3
  const int wr   = wv >> 1;           // wave row 0..1
  const int wc   = wv & 1;            // wave col 0..1
  const int half = lane >> 4;
  const int l16  = lane & 15;

  const int m0 = blockIdx.y * BM;
  const int n0 = blockIdx.x * BN;

  v8f acc00 = {}; v8f acc01 = {}; v8f acc10 = {}; v8f acc11 = {};

  const int ar = tid >> 1;            // A loader: row 0..63
  const int ac = (tid & 1) * 8;       // A loader: col 0 or 8
  const int bk = tid & 15;            // B loader: k row 0..15
  const int bn = (tid >> 4) * 8;      // B loader: n base 0..56

  for (int k0 = 0; k0 < K; k0 += BK) {
    __syncthreads();
    // ---- stage A (+A2) tile 64x16 into LDS ----
    {
      const float* ag = A + (long)(m0 + ar) * K + k0 + ac;
      v4f a0 = *(const v4f*)(ag);
      v4f a1 = *(const v4f*)(ag + 4);
      if (A2) {
        const float* a2g = A2 + (long)(m0 + ar) * K + k0 + ac;
        a0 += *(const v4f*)(a2g);
        a1 += *(const v4f*)(a2g + 4);
      }
      *(v4f*)&Al[ar * LSTR + ac]     = a0;
      *(v4f*)&Al[ar * LSTR + ac + 4] = a1;
    }
    // ---- stage B tile 16x64 transposed into LDS ----
    {
      const float* bg = B + (long)(k0 + bk) * N + n0 + bn;
      v4f b0 = *(const v4f*)(bg);
      v4f b1 = *(const v4f*)(bg + 4);
      #pragma unroll
      for (int i = 0; i < 4; ++i) Bl[(bn + i) * LSTR + bk]     = b0[i];
      #pragma unroll
      for (int i = 0; i < 4; ++i) Bl[(bn + 4 + i) * LSTR + bk] = b1[i];
    }
    __syncthreads();
    // ---- 4 k-steps of 2x2 WMMA tiles ----
    #pragma unroll
    for (int ks = 0; ks < 4; ++ks) {
      const int kk = ks * 4 + half * 2;
      v2f a0 = *(const v2f*)&Al[(wr * 32 + l16)      * LSTR + kk];
      v2f a1 = *(const v2f*)&Al[(wr * 32 + 16 + l16) * LSTR + kk];
      v2f b0 = *(const v2f*)&Bl[(wc * 32 + l16)      * LSTR + kk];
      v2f b1 = *(const v2f*)&Bl[(wc * 32 + 16 + l16) * LSTR + kk];
      acc00 = wmma4(a0, b0, acc00);
      acc01 = wmma4(a0, b1, acc01);
      acc10 = wmma4(a1, b0, acc10);
      acc11 = wmma4(a1, b1, acc11);
    }
  }

  // ---- epilogue: bias -> act -> (mul) -> store ----
  #pragma unroll
  for (int mt = 0; mt < 2; ++mt) {
    #pragma unroll
    for (int nt = 0; nt < 2; ++nt) {
      v8f acc = (mt == 0) ? ((nt == 0) ? acc00 : acc01)
                          : ((nt == 0) ? acc10 : acc11);
      const int col = n0 + wc * 32 + nt * 16 + l16;
      const float bb = bias ? bias[col] : 0.0f;
      #pragma unroll
      for (int v = 0; v < 8; ++v) {
        const int row = m0 + wr * 32 + mt * 16 + half * 8 + v;
        float val = acc[v] + bb;
        if      (act == ACT_TANH) val = tanhf(val);
        else if (act == ACT_RELU) val = fmaxf(val, 0.0f);
        else if (act == ACT_SIGM) val = 1.0f / (1.0f + __expf(-val));
        if (MulP) val *= MulP[(long)row * N + col];
        C[(long)row * N + col] = val;
      }
    }
  }
}

// ---------------------------------------------------------------------------
// Flash attention, per-head dim 128. Q,K,V stored [H][N][128].
// Block: 128 thr (4 waves), each wave owns 16 query rows; grid (N/64, H).
// Ctx written [N][H*128] (heads concatenated) for the Wattn GEMM.
// ---------------------------------------------------------------------------
#define KSTR 132  // padded K/V tile row stride
#define PSTR 66   // padded P tile row stride

__global__ __launch_bounds__(128)
void attn_flash(const float* Q, const float* Km, const float* V, float* Ctx,
                int N, int H)
{
  __shared__ float Kt[64 * KSTR];
  __shared__ float Vt[64 * KSTR];
  __shared__ float Pl[4 * 16 * PSTR];

  const int h    = blockIdx.y;
  const int qt   = blockIdx.x;
  const int tid  = threadIdx.x;
  const int lane = tid & 31;
  const int wv   = tid >> 5;
  const int half = lane >> 4;
  const int l16  = lane & 15;

  const long hoff = (long)h * N * 128;
  const float* Qh = Q  + hoff;
  const float* Kh = Km + hoff;
  const float* Vh = V  + hoff;

  const int qrow0 = qt * 64 + wv * 16;

  // Q fragments for all 32 k-chunks (K-dim = 128)
  v2f qf[32];
  #pragma unroll
  for (int kc = 0; kc < 32; ++kc)
    qf[kc] = *(const v2f*)&Qh[(long)(qrow0 + l16) * 128 + kc * 4 + half * 2];

  v8f o[8];
  #pragma unroll
  for (int i = 0; i < 8; ++i) o[i] = (v8f){};
  float mrow[8], srow[8];
  #pragma unroll
  for (int v = 0; v < 8; ++v) { mrow[v] = -3.0e38f; srow[v] = 0.0f; }

  const float scale = 0.08838834764831845f;  // 1/sqrt(128)
  const int ldrow = tid >> 1;
  const int ldcol = (tid & 1) * 64;
  float* Pw = &Pl[wv * 16 * PSTR];

  for (int kt0 = 0; kt0 < N; kt0 += 64) {
    // ---- stage K,V tiles (64x128) into padded LDS ----
    {
      const float* kg = &Kh[(long)(kt0 + ldrow) * 128 + ldcol];
      const float* vg = &Vh[(long)(kt0 + ldrow) * 128 + ldcol];
      float* kl = &Kt[ldrow * KSTR + ldcol];
      float* vl = &Vt[ldrow * KSTR + ldcol];
      #pragma unroll
      for (int i = 0; i < 16; ++i) {
        *(v4f*)(kl + i * 4) = *(const v4f*)(kg + i * 4);
        *(v4f*)(vl + i * 4) = *(const v4f*)(vg + i * 4);
      }
    }
    __syncthreads();

    // ---- S = Q @ K^T  (16 queries x 64 keys per wave) ----
    v8f s[4];
    #pragma unroll
    for (int nt = 0; nt < 4; ++nt) s[nt] = (v8f){};
    #pragma unroll
    for (int kc = 0; kc < 32; ++kc) {
      const int kk = kc * 4 + half * 2;
      #pragma unroll
      for (int nt = 0; nt < 4; ++nt) {
        v2f b = *(const v2f*)&Kt[(nt * 16 + l16) * KSTR + kk];
        s[nt] = wmma4(qf[kc], b, s[nt]);
      }
    }

    // ---- online softmax (rows live in a 16-lane half; xor<16 stays inside) ----
    #pragma unroll
    for (int v = 0; v < 8; ++v) {
      float mx = -3.0e38f;
      #pragma unroll
      for (int nt = 0; nt < 4; ++nt) mx = fmaxf(mx, s[nt][v] * scale);
      for (int d = 1; d < 16; d <<= 1) mx = fmaxf(mx, __shfl_xor(mx, d, 32));
      const float mnew  = fmaxf(mrow[v], mx);
      const float alpha = __expf(mrow[v] - mnew);
      float rs = 0.0f;
      #pragma unroll
      for (int nt = 0; nt < 4; ++nt) {
        float p = __expf(s[nt][v] * scale - mnew);
        s[nt][v] = p;
        rs += p;
      }
      for (int d = 1; d < 16; d <<= 1) rs += __shfl_xor(rs, d, 32);
      srow[v] = srow[v] * alpha + rs;
      mrow[v] = mnew;
      #pragma unroll
      for (int nt2 = 0; nt2 < 8; ++nt2) o[nt2][v] *= alpha;
    }

    // ---- transpose P (C layout -> A layout) through per-wave LDS ----
    #pragma unroll
    for (int v = 0; v < 8; ++v)
      #pragma unroll
      for (int nt = 0; nt < 4; ++nt)
        Pw[(half * 8 + v) * PSTR + nt * 16 + l16] = s[nt][v];

    // ---- O += P @ V ----
    #pragma unroll
    for (int kc = 0; kc < 16; ++kc) {
      const int kk = kc * 4 + half * 2;
      v2f a = *(const v2f*)&Pw[l16 * PSTR + kk];
      #pragma unroll
      for (int nt = 0; nt < 8; ++nt) {
        v2f b;
        b.x = Vt[(kk)     * KSTR + nt * 16 + l16];
        b.y = Vt[(kk + 1) * KSTR + nt * 16 + l16];
        o[nt] = wmma4(a, b, o[nt]);
      }
    }
    __syncthreads();
  }

  // ---- write ctx[n][h*128+d] = O / rowsum ----
  #pragma unroll
  for (int nt = 0; nt < 8; ++nt)
    #pragma unroll
    for (int v = 0; v < 8; ++v) {
      const int row = qrow0 + half * 8 + v;
      const int col = h * 128 + nt * 16 + l16;
      Ctx[(long)row * (H * 128) + col] = o[nt][v] / srow[v];
    }
}

// ---------------------------------------------------------------------------
// Final 512 -> 8 layer with tanh (tiny; plain VALU)
// ---------------------------------------------------------------------------
__global__ void out_layer(const float* X, const float* W, const float* b,
                          float* Out, int rows)
{
  int idx = blockIdx.x * blockDim.x + threadIdx.x;
  if (idx >= rows * 8) return;
  int n = idx >> 3, oc = idx & 7;
  const float* x = X + (long)n * 512;
  float acc = b[oc];
  for (int i = 0; i < 512; ++i) acc = fmaf(x[i], W[i * 8 + oc], acc);
  Out[idx] = tanhf(acc);
}

// ---------------------------------------------------------------------------
static inline void gemm(hipStream_t s, const float* A, const float* A2,
                        const float* B, const float* bias, const float* Mul,
                        float* C, int M, int N, int K, int batch,
                        long sA, long sB, long sC, int act)
{
  dim3 g(N / 64, M / 64, batch), blk(128);
  gemm_wmma_f32<<<g, blk, 0, s>>>(A, A2, B, bias, Mul, C, M, N, K, sA, sB, sC, act);
}

extern "C" void kernel_launch(void* const* d_in, const int* in_sizes, int n_in,
                              void* d_out, int out_size, void* d_ws, size_t ws_size,
                              hipStream_t stream)
{
  (void)in_sizes; (void)n_in; (void)out_size; (void)ws_size;
  const int N = 4096, HN = 16384;

  // Input order: setup_inputs() dict insertion order, params flattened recursively.
  const float* input = (const float*)d_in[0];
  const float* pay0  = (const float*)d_in[1];
  const float* pay1  = (const float*)d_in[2];
  const float* Wq    = (const float*)d_in[3];
  const float* Wk0   = (const float*)d_in[4];
  const float* Wk1   = (const float*)d_in[5];
  const float* Wv0   = (const float*)d_in[6];
  const float* Wv1   = (const float*)d_in[7];
  const float* Wattn = (const float*)d_in[8];
  const float* lin0W = (const float*)d_in[9];
  const float* lin0b = (const float*)d_in[10];
  const float* lin1W = (const float*)d_in[11];
  const float* lin1b = (const float*)d_in[12];
  const float* lin2W = (const float*)d_in[13];
  const float* lin2b = (const float*)d_in[14];
  const float* outW  = (const float*)d_in[15];
  const float* outb  = (const float*)d_in[16];
  const float* kn0W1 = (const float*)d_in[17];
  const float* kn0b1 = (const float*)d_in[18];
  const float* kn0W2 = (const float*)d_in[19];
  const float* kn0b2 = (const float*)d_in[20];
  const float* kn1W1 = (const float*)d_in[21];
  const float* kn1b1 = (const float*)d_in[22];
  const float* kn1W2 = (const float*)d_in[23];
  const float* kn1b2 = (const float*)d_in[24];
  const float* vn0W1 = (const float*)d_in[25];
  const float* vn0b1 = (const float*)d_in[26];
  const float* vn0W2 = (const float*)d_in[27];
  const float* vn0b2 = (const float*)d_in[28];
  const float* vn1W1 = (const float*)d_in[29];
  const float* vn1b1 = (const float*)d_in[30];
  const float* vn1W2 = (const float*)d_in[31];
  const float* vn1b2 = (const float*)d_in[32];
  const float* koW1  = (const float*)d_in[33];
  const float* kob1  = (const float*)d_in[34];
  const float* koW2  = (const float*)d_in[35];
  const float* kob2  = (const float*)d_in[36];
  const float* voW1  = (const float*)d_in[37];
  const float* vob1  = (const float*)d_in[38];
  const float* voW2  = (const float*)d_in[39];
  const float* vob2  = (const float*)d_in[40];

  // workspace layout (floats)
  float* ws  = (float*)d_ws;
  float* x0  = ws;                    // [4096,512]
  float* x1  = x0  + 2097152;
  float* q   = x1  + 2097152;         // [H][4096][128]
  float* P   = q   + 2097152;         // [16384,512] product buffer (k then v)
  float* scr = P   + 8388608;         // pre@0 / h1@+2M / keyout-hidden@0
  float* kb  = scr + 8388608;         // [H][4096][128]
  float* vb  = kb  + 2097152;
  float* ctx = vb  + 2097152;         // [4096,512]
  float* inj = ctx + 2097152;         // [4096,512]
  float* pre = scr;
  float* h1  = scr + 2097152;
  float* kh  = scr;                   // reuses pre/h1 region (dead by then)

  // x = tanh(input @ W0 + b0)
  gemm(stream, input, 0, lin0W, lin0b, 0, x0, N, 512, 512, 1, 0, 0, 0, ACT_TANH);

  for (int l = 1; l <= 2; ++l) {
    const float* xin  = (l == 1) ? x0 : x1;
    float*       xout = (l == 1) ? x1 : x0;
    const float* Wl   = (l == 1) ? lin1W : lin2W;
    const float* bl   = (l == 1) ? lin1b : lin2b;

    // q = x @ Wq[h]   (batched over heads)
    gemm(stream, xin, 0, Wq, 0, 0, q, N, 128, 512, 4, 0, 512 * 128, (long)N * 128, ACT_NONE);

    // ---- key chain -> kb ----
    gemm(stream, pay0, 0, Wk0, 0, 0, pre, N, 128, 512, 4, 0, 512 * 128, (long)N * 128, ACT_NONE);
    gemm(stream, pre, 0, kn0W1, kn0b1, 0, h1, HN, 256, 128, 1, 0, 0, 0, ACT_RELU);
    gemm(stream, h1,  0, kn0W2, kn0b2, 0, P,  HN, 512, 256, 1, 0, 0, 0, ACT_NONE);
    gemm(stream, pay1, 0, Wk1, 0, 0, pre, N, 64, 256, 4, 0, 256 * 64, (long)N * 64, ACT_NONE);
    gemm(stream, pre, 0, kn1W1, kn1b1, 0, h1, HN, 256, 64, 1, 0, 0, 0, ACT_RELU);
    gemm(stream, h1,  0, kn1W2, kn1b2, P, P,  HN, 512, 256, 1, 0, 0, 0, ACT_NONE); // *= kenc0
    gemm(stream, P,  0, koW1, kob1, 0, kh, HN, 512, 512, 1, 0, 0, 0, ACT_RELU);
    gemm(stream, kh, 0, koW2, kob2, 0, kb, HN, 128, 512, 1, 0, 0, 0, ACT_SIGM);

    // ---- value chain -> vb ----
    gemm(stream, pay0, 0, Wv0, 0, 0, pre, N, 128, 512, 4, 0, 512 * 128, (long)N * 128, ACT_NONE);
    gemm(stream, pre, 0, vn0W1, vn0b1, 0, h1, HN, 256, 128, 1, 0, 0, 0, ACT_RELU);
    gemm(stream, h1,  0, vn0W2, vn0b2, 0, P,  HN, 512, 256, 1, 0, 0, 0, ACT_NONE);
    gemm(stream, pay1, 0, Wv1, 0, 0, pre, N, 64, 256, 4, 0, 256 * 64, (long)N * 64, ACT_NONE);
    gemm(stream, pre, 0, vn1W1, vn1b1, 0, h1, HN, 256, 64, 1, 0, 0, 0, ACT_RELU);
    gemm(stream, h1,  0, vn1W2, vn1b2, P, P,  HN, 512, 256, 1, 0, 0, 0, ACT_NONE); // *= venc0
    gemm(stream, P,  0, voW1, vob1, 0, kh, HN, 512, 512, 1, 0, 0, 0, ACT_RELU);
    gemm(stream, kh, 0, voW2, vob2, 0, vb, HN, 128, 512, 1, 0, 0, 0, ACT_SIGM);

    // ---- attention + output projection + residual mixing ----
    attn_flash<<<dim3(N / 64, 4), 128, 0, stream>>>(q, kb, vb, ctx, N, 4);
    gemm(stream, ctx, 0, Wattn, 0, 0, inj, N, 512, 512, 1, 0, 0, 0, ACT_NONE);
    gemm(stream, xin, inj, Wl, bl, 0, xout, N, 512, 512, 1, 0, 0, 0, ACT_NONE);
  }

  // out = tanh(x @ Wo + bo)   (x ends in x0 after layer 2)
  out_layer<<<(N * 8 + 255) / 256, 256, 0, stream>>>(x0, outW, outb, (float*)d_out, N);
}